// FlowPPartial_9655086481807
// MI455X (gfx1250) — compile-verified
//
#include <hip/hip_runtime.h>
#include <stdint.h>

// Problem geometry (fixed by the reference's setup_inputs)
constexpr int B = 8, S = 128, R = 192, C = 192;
constexpr int TJ = 16;                 // rows (R-axis) per workgroup strip
constexpr int NSTRIP = R / TJ;         // 12 strips per (b, i) plane

typedef unsigned int v4u __attribute__((ext_vector_type(4)));
typedef int          v8i __attribute__((ext_vector_type(8)));
typedef int          v4i __attribute__((ext_vector_type(4)));

__device__ __forceinline__ int iclamp(int v, int lo, int hi) {
    return v < lo ? lo : (v > hi ? hi : v);
}

// Issue a 1-D TDM copy of nElem floats from global `src` into LDS at byte
// offset `ldsByteOff`. Descriptor layout per CDNA5 ISA sec 8 (D# groups 0-3).
// Tracked by TENSORcnt; caller must s_wait_tensorcnt.
__device__ __forceinline__ void tdm_copy_1d(const float* src, uint32_t ldsByteOff,
                                            uint32_t nElem) {
    uint64_t ga = (uint64_t)(uintptr_t)src;
    v4u g0;
    g0.x = 1u;                                         // count=1, user descriptor
    g0.y = ldsByteOff;                                 // lds_addr (bytes)
    g0.z = (uint32_t)(ga & 0xFFFFFFFFu);               // global_addr[31:0]
    g0.w = (uint32_t)((ga >> 32) & 0x01FFFFFFu)        // global_addr[56:32]
         | (2u << 30);                                 // type = 2 ("image")
    v8i g1;
    g1[0] = (int)(2u << 16);                           // data_size = 4 bytes; mask=0
    g1[1] = (int)((nElem & 0xFFFFu) << 16);            // tensor_dim0[15:0]
    g1[2] = (int)(((nElem >> 16) & 0xFFFFu)            // tensor_dim0[31:16]
         | (1u << 16));                                // tensor_dim1 = 1 (lo bits)
    g1[3] = (int)((nElem & 0xFFFFu) << 16);            // tile_dim0 = nElem (16-bit)
    g1[4] = (int)1u;                                   // tile_dim1 = 1, tile_dim2 = 0
    g1[5] = (int)nElem;                                // tensor_dim0_stride[31:0]
    g1[6] = 0;                                         // stride hi / dim1_stride lo
    g1[7] = 0;
    v4i g2 = {0, 0, 0, 0};                             // unused (<=2D tensor)
    v4i g3 = {0, 0, 0, 0};
    v8i g4 = {0, 0, 0, 0, 0, 0, 0, 0};                 // clang-23 extra operand
    __builtin_amdgcn_tensor_load_to_lds(g0, g1, g2, g3, g4, /*cpol=*/0);
}

__global__ __launch_bounds__(192)
void flowp_stencil_kernel(const float* __restrict__ P,
                          const float* __restrict__ D,
                          float* __restrict__ out) {
    // LDS staging: center plane strip (TJ+2 rows), +/- S-plane strips (TJ rows)
    __shared__ float sC[(TJ + 2) * C];
    __shared__ float sM[TJ * C];
    __shared__ float sP[TJ * C];

    const int blk = blockIdx.x;
    const int js  = blk % NSTRIP;
    const int i   = (blk / NSTRIP) % S;
    const int b   = blk / (NSTRIP * S);
    const int j0  = js * TJ;

    // Clamped S-axis plane indices (Neumann bc + edge pad => clamp to [1, S-2])
    const int exc = iclamp(i,     1, S - 2);
    const int exm = iclamp(i - 1, 1, S - 2);
    const int exq = iclamp(i + 1, 1, S - 2);

    // Row ranges needed from each plane (R-axis clamp to [1, R-2])
    const int rlo  = iclamp(j0 - 1,      1, R - 2);
    const int rhi  = iclamp(j0 + TJ,     1, R - 2);
    const int nCc  = rhi - rlo + 1;                    // <= TJ+2 rows
    const int rlo2 = iclamp(j0,          1, R - 2);
    const int rhi2 = iclamp(j0 + TJ - 1, 1, R - 2);
    const int nMm  = rhi2 - rlo2 + 1;                  // <= TJ rows

    const size_t plane = (size_t)R * C;
    const float* baseB = P + (size_t)b * S * plane;

    // One wave drives the three TDM strip loads, waits, then block barrier.
    if (threadIdx.x < 32) {
        tdm_copy_1d(baseB + (size_t)exc * plane + (size_t)rlo  * C,
                    (uint32_t)(uintptr_t)sC, (uint32_t)(nCc * C));
        tdm_copy_1d(baseB + (size_t)exm * plane + (size_t)rlo2 * C,
                    (uint32_t)(uintptr_t)sM, (uint32_t)(nMm * C));
        tdm_copy_1d(baseB + (size_t)exq * plane + (size_t)rlo2 * C,
                    (uint32_t)(uintptr_t)sP, (uint32_t)(nMm * C));
        __builtin_amdgcn_s_wait_tensorcnt(0);
    }
    __syncthreads();

    // Per-thread column; C-axis clamps are loop-invariant.
    const int k   = threadIdx.x;                       // 0..191
    const int kz  = iclamp(k,     1, C - 2);
    const int kzm = iclamp(k - 1, 1, C - 2);
    const int kzp = iclamp(k + 1, 1, C - 2);

    // SPACING = (2.0, 1.5, 1.5) on axes (S, R, C)
    const float inv2dx = 0.25f;            // 1/(2*2.0)
    const float inv2dy = 1.0f / 3.0f;      // 1/(2*1.5)
    const float inv2dz = 1.0f / 3.0f;
    const float invdx2 = 0.25f;            // 1/(2.0^2)
    const float invdy2 = 1.0f / 2.25f;     // 1/(1.5^2)
    const float invdz2 = 1.0f / 2.25f;

    const size_t rowBase = ((size_t)b * S + (size_t)i) * plane + (size_t)j0 * C + (size_t)k;

    #pragma unroll 4
    for (int jj = 0; jj < TJ; ++jj) {
        const int j   = j0 + jj;
        const int ey  = iclamp(j,     1, R - 2);
        const int eym = iclamp(j - 1, 1, R - 2);
        const int eyp = iclamp(j + 1, 1, R - 2);

        const float* rowC = &sC[(ey - rlo) * C];
        const float p0  = rowC[kz];
        const float pzm = rowC[kzm];
        const float pzp = rowC[kzp];
        const float pym = sC[(eym - rlo)  * C + kz];
        const float pyp = sC[(eyp - rlo)  * C + kz];
        const float pxm = sM[(ey  - rlo2) * C + kz];
        const float pxp = sP[(ey  - rlo2) * C + kz];

        const float cx = (pxp - pxm) * inv2dx;
        const float cy = (pyp - pym) * inv2dy;
        const float cz = (pzp - pzm) * inv2dz;
        const float F  = cx * cx + cy * cy + cz * cz;
        const float G  = (pxp - 2.0f * p0 + pxm) * invdx2
                       + (pyp - 2.0f * p0 + pym) * invdy2
                       + (pzp - 2.0f * p0 + pzm) * invdz2;

        // D is single-use, out is write-once: keep both out of L2's way.
        const float d = __builtin_nontemporal_load(&D[rowBase + (size_t)jj * C]);
        __builtin_nontemporal_store(-0.5f * F - d * G, &out[rowBase + (size_t)jj * C]);
    }
}

extern "C" void kernel_launch(void* const* d_in, const int* in_sizes, int n_in,
                              void* d_out, int out_size, void* d_ws, size_t ws_size,
                              hipStream_t stream) {
    // setup_inputs order: t (unused), batch_P, D
    const float* P = (const float*)d_in[1];
    const float* D = (const float*)d_in[2];
    float* out     = (float*)d_out;

    dim3 grid(B * S * NSTRIP);   // 12288 blocks
    dim3 block(192);             // 6 wave32 = one k-row
    flowp_stencil_kernel<<<grid, block, 0, stream>>>(P, D, out);
}